// GAUPostProcessor_71081708749425
// MI455X (gfx1250) — compile-verified
//
#include <hip/hip_runtime.h>
#include <math.h>

// ---------------- constants (mirror the reference) ----------------
#define NIMG   4
#define NCLS   80
#define TOPK   500
#define TOPPAD 512
#define KTOT   1500         // 3 levels * 500
#define POSTK  100
#define CAP    131072       // per-(image,level) raw candidate capacity
#define SELCAP 4096
#define IMGW_F 2048.0f
#define IMGH_F 1280.0f
#define SIGMA2_F 0.125f     // (0.25*sqrt(2))^2
#define CLS_TH_LOGIT (-2.9444389791664403f)  // ln(0.05/0.95)
#define NMS_TH 0.5f
#define SCAN_TH 16          // strip height
#define SCAN_B  256

// ---------------- small device helpers ----------------
__device__ __forceinline__ float sigm(float x) { return 1.0f / (1.0f + __expf(-x)); }
// stable softplus(z) = log(1+e^z); L = SIGMA2 * softplus(-g) == -SIGMA2*log(sigmoid(g))
__device__ __forceinline__ float softplusf(float z) {
    return fmaxf(z, 0.0f) + log1pf(__expf(-fabsf(z)));
}
// Low 32 bits of a generic LDS pointer are the LDS byte address (ISA 10.2 aperture table).
__device__ __forceinline__ unsigned lds_byte_addr(const void* p) {
    return (unsigned)(unsigned long long)p;
}
// CDNA5 async global->LDS copy, 16B per lane, GVS addressing. Tracked by ASYNCcnt.
__device__ __forceinline__ void async_g2l_b128(unsigned ldsAddr, const void* gbase, unsigned voff) {
    asm volatile("global_load_async_to_lds_b128 %0, %1, %2"
                 :: "v"(ldsAddr), "v"(voff), "s"(gbase)
                 : "memory");
}
__device__ __forceinline__ void wait_async0() {
#if __has_builtin(__builtin_amdgcn_s_wait_asynccnt)
    __builtin_amdgcn_s_wait_asynccnt(0);
#else
    asm volatile("s_wait_asynccnt 0" ::: "memory");
#endif
}

// Descending bitonic sort of (key,payload) in LDS. Ties: smaller payload first
// (matches jnp argsort/top_k stable tie-breaking by index).
__device__ void bitonic_desc(float* k, int* p, int M, int tid, int nt) {
    for (int size = 2; size <= M; size <<= 1) {
        for (int stride = size >> 1; stride > 0; stride >>= 1) {
            __syncthreads();
            for (int t = tid; t < M; t += nt) {
                int j = t ^ stride;
                if (j > t) {
                    float ka = k[t], kb = k[j];
                    int   pa = p[t], pb = p[j];
                    bool aFirst = (ka > kb) || (ka == kb && pa < pb); // desired order: a before b
                    bool dirDesc = ((t & size) == 0);
                    bool inOrder = dirDesc ? aFirst : !aFirst;
                    if (!inOrder) { k[t] = kb; k[j] = ka; p[t] = pb; p[j] = pa; }
                }
            }
        }
    }
    __syncthreads();
}

// ---------------- kernel 0: zero candidate counters ----------------
__global__ void init_kernel(int* counters) {
    if (threadIdx.x < NIMG * 3) counters[threadIdx.x] = 0;
}

// ---------------- kernel 1: streaming scan (memory-bound, 137.6 MB) ----------------
// Grid: (numStrips, C, N). Each block stages an 18-row halo strip of gau logits
// into LDS via the CDNA5 async data-mover path, tests 3x3 local max + class
// threshold, and compacts candidates (score, flat_idx) with an atomic counter.
// Templated on (H, W): W is a power of two -> div/mod become shifts, LDS tile
// sized per level, async-issue loops fully unrolled.
template <int H, int W>
__global__ void __launch_bounds__(SCAN_B)
scan_kernel(const float* __restrict__ cls, const float* __restrict__ gau,
            int lvl,
            float* __restrict__ candS, int* __restrict__ candI, int* __restrict__ counters)
{
    __shared__ __align__(16) float tile[(SCAN_TH + 2) * W];

    const int strip = blockIdx.x, c = blockIdx.y, n = blockIdx.z;
    const int tid = threadIdx.x;
    const int y0 = strip * SCAN_TH;
    constexpr int HW = H * W;

    const float* gp = gau + ((size_t)n * NCLS + c) * (size_t)HW;
    const float* cp = cls + ((size_t)n * NCLS + c) * (size_t)HW;

    // Halo rows [max(y0-1,0) .. min(y0+TH, H-1)] are contiguous in memory.
    const int gstart = (y0 > 0) ? (y0 - 1) : 0;
    int gend = y0 + SCAN_TH; if (gend > H - 1) gend = H - 1;
    const int rows = gend - gstart + 1;
    const int ldsRow0 = (y0 > 0) ? 0 : 1;

    const unsigned nbytes = (unsigned)(rows * W) * 4u;
    const char* gbase = (const char*)(gp + (size_t)gstart * W);
    const unsigned ldsBase = lds_byte_addr(tile) + (unsigned)(ldsRow0 * W * 4);

    for (unsigned off = (unsigned)tid * 16u; off < nbytes; off += SCAN_B * 16u)
        async_g2l_b128(ldsBase + off, gbase, off);
    wait_async0();
    __syncthreads();

    for (int pIt = tid; pIt < SCAN_TH * W; pIt += SCAN_B) {
        const int dy = pIt / W, x = pIt & (W - 1);   // W is a power of two
        const int y = y0 + dy;
        if (y < 1 || y > H - 2 || x < 1 || x > W - 2) continue;   // interior mask
        const int lr = y - y0 + 1;                                // lds row of y
        const float* t0 = &tile[(lr - 1) * W + x];
        const float* t1 = &tile[lr * W + x];
        const float* t2 = &tile[(lr + 1) * W + x];
        const float g = t1[0];
        float m = g;
        m = fmaxf(m, fmaxf(t0[-1], fmaxf(t0[0], t0[1])));
        m = fmaxf(m, fmaxf(t1[-1], t1[1]));
        m = fmaxf(m, fmaxf(t2[-1], fmaxf(t2[0], t2[1])));
        if (g != m) continue;                       // sigmoid monotonic: logit max == prob max
        const float cl = __builtin_nontemporal_load(&cp[y * W + x]);
        if (!(cl > CLS_TH_LOGIT)) continue;         // box_p > 0.05
        const float score = sqrtf(sigm(g) * sigm(cl));
        if (!(score > 0.0f)) continue;
        const int pos = atomicAdd(&counters[n * 3 + lvl], 1);
        if (pos < CAP) {
            const size_t base = (size_t)(n * 3 + lvl) * CAP;
            candS[base + pos] = score;
            candI[base + pos] = c * HW + y * W + x;
        }
    }
}

// ---------------- kernel 2: exact-ish top-500 per (image, level) ----------------
__global__ void __launch_bounds__(1024)
select_kernel(const float* __restrict__ candS, const int* __restrict__ candI,
              const int* __restrict__ counters,
              float* __restrict__ topS, int* __restrict__ topI)
{
    __shared__ int hist[2048];
    __shared__ float selS[SELCAP];
    __shared__ int selI[SELCAP];
    __shared__ int selCnt, thrBin;

    const int b = blockIdx.x;        // n*3 + lvl
    const int tid = threadIdx.x;
    int cnt = counters[b]; if (cnt > CAP) cnt = CAP;
    const size_t base = (size_t)b * CAP;

    for (int i = tid; i < 2048; i += 1024) hist[i] = 0;
    if (tid == 0) { selCnt = 0; thrBin = 0; }
    __syncthreads();

    // Monotonic float-bits -> 2048-bin histogram (scores in (0,1])
    for (int i = tid; i < cnt; i += 1024) {
        int bin = ((int)__float_as_uint(candS[base + i]) - 0x3C000000) >> 15;
        bin = bin < 0 ? 0 : (bin > 2047 ? 2047 : bin);
        atomicAdd(&hist[bin], 1);
    }
    __syncthreads();
    if (tid == 0) {
        int acc = 0;
        for (int i = 2047; i >= 0; --i) { acc += hist[i]; if (acc >= TOPK) { thrBin = i; break; } }
    }
    __syncthreads();
    const int T = thrBin;
    for (int i = tid; i < cnt; i += 1024) {
        const float s = candS[base + i];
        int bin = ((int)__float_as_uint(s) - 0x3C000000) >> 15;
        bin = bin < 0 ? 0 : (bin > 2047 ? 2047 : bin);
        if (bin >= T) {
            const int pos = atomicAdd(&selCnt, 1);
            if (pos < SELCAP) { selS[pos] = s; selI[pos] = candI[base + i]; }
        }
    }
    __syncthreads();
    int cN = selCnt; if (cN > SELCAP) cN = SELCAP;
    for (int i = tid; i < SELCAP; i += 1024)
        if (i >= cN) { selS[i] = -1.0e30f; selI[i] = 0x7FFFFFFF; }

    bitonic_desc(selS, selI, SELCAP, tid, 1024);

    for (int k = tid; k < TOPK; k += 1024) {
        const bool ok = (k < cN) && (selS[k] > 0.0f);
        topS[b * TOPPAD + k] = ok ? selS[k] : 0.0f;   // reference top_k pads with zeros
        topI[b * TOPPAD + k] = ok ? selI[k] : 0;
    }
}

// ---------------- kernel 3: quadratic refinement ("solve", a=b=1) ----------------
// Note: the reference's bad = (err>1.2)&(err<1/1.2) is identically false, so
// bx == s1 always; the a=2 solve is dead code and omitted.
__global__ void __launch_bounds__(256)
solve_kernel(const float* __restrict__ g0, const float* __restrict__ g1, const float* __restrict__ g2,
             const float* __restrict__ topS, const int* __restrict__ topI,
             float* __restrict__ boxes, float* __restrict__ rsc,
             int* __restrict__ rlab, int* __restrict__ rval)
{
    const int gid = blockIdx.x * blockDim.x + threadIdx.x;
    if (gid >= NIMG * KTOT) return;
    const int n = gid / KTOT;
    const int r = gid % KTOT;
    const int lvl = r / TOPK, slot = r % TOPK;

    const int Hs[3] = {160, 80, 40}, Ws[3] = {256, 128, 64}, st[3] = {8, 16, 32};
    const float* gptr = (lvl == 0) ? g0 : ((lvl == 1) ? g1 : g2);
    const int H = Hs[lvl], W = Ws[lvl], HW = H * W;
    const float step = (float)st[lvl];

    const float s = topS[(n * 3 + lvl) * TOPPAD + slot];
    int idx = topI[(n * 3 + lvl) * TOPPAD + slot];
    if (idx < 0 || idx >= NCLS * HW) idx = 0;
    const int c = idx / HW, rem = idx % HW, y = rem / W, x = rem % W;
    const float* gp = gptr + ((size_t)n * NCLS + c) * (size_t)HW;

    auto Lat = [&](int dy, int dx) -> float {
        int yy = y + dy; yy = yy < 0 ? 0 : (yy > H - 1 ? H - 1 : yy);
        int xx = x + dx; xx = xx < 0 ? 0 : (xx > W - 1 ? W - 1 : xx);
        return SIGMA2_F * softplusf(-gp[yy * W + xx]);   // -log(sigmoid)*SIGMA2
    };
    const float l0 = Lat(0, 0);
    const float lxp = Lat(0, 1), lxm = Lat(0, -1);
    const float lyp = Lat(1, 0), lym = Lat(-1, 0);
    const float eps = 1e-8f;
    const float Ax = (lxp + lxm - 2.0f * l0) * 0.5f;     // a=1
    const float Ay = (lyp + lym - 2.0f * l0) * 0.5f;     // b=1
    const float Axs = Ax > eps ? Ax : 1.0f;
    const float Ays = Ay > eps ? Ay : 1.0f;
    const float mux = (float)x - (lxp - lxm) / (4.0f * Axs);
    const float muy = (float)y - (lyp - lym) / (4.0f * Ays);
    const float wb = (Ax > eps ? rsqrtf(2.0f * Axs) : 0.0f) * step;
    const float hb = (Ay > eps ? rsqrtf(2.0f * Ays) : 0.0f) * step;
    const float half = (step - 1.0f) * 0.5f;
    float x1 = (mux * step - 0.5f * wb) + half;
    float y1 = (muy * step - 0.5f * hb) + half;
    const int valid = (s > 0.0f) && (wb > 0.0f) && (hb > 0.0f);
    float x2 = x1 + wb - 1.0f, y2 = y1 + hb - 1.0f;
    x1 = fminf(fmaxf(x1, 0.0f), IMGW_F - 1.0f);
    x2 = fminf(fmaxf(x2, 0.0f), IMGW_F - 1.0f);
    y1 = fminf(fmaxf(y1, 0.0f), IMGH_F - 1.0f);
    y2 = fminf(fmaxf(y2, 0.0f), IMGH_F - 1.0f);

    const size_t o = (size_t)n * KTOT + r;
    boxes[o * 4 + 0] = x1; boxes[o * 4 + 1] = y1;
    boxes[o * 4 + 2] = x2; boxes[o * 4 + 3] = y2;
    rsc[o] = s; rlab[o] = c + 1; rval[o] = valid;
}

// ---------------- kernel 4: per-image sort + greedy NMS + top-100 ----------------
__global__ void __launch_bounds__(1024)
nms_kernel(const float* __restrict__ boxes, const float* __restrict__ rsc,
           const int* __restrict__ rlab, const int* __restrict__ rval,
           float* __restrict__ out)
{
    __shared__ float skey[2048];
    __shared__ int spid[2048];                 // reused as outpos after gather
    __shared__ float sx1[1536], sy1[1536], sx2[1536], sy2[1536], sar[1536];
    __shared__ int slab[1536], keep[1536];

    const int n = blockIdx.x, tid = threadIdx.x;
    const size_t base = (size_t)n * KTOT;

    for (int j = tid; j < 2048; j += 1024) {
        if (j < KTOT) {
            float s = rsc[base + j];
            if (!rval[base + j]) s = -1.0f;    // invalid -> score -1 (reference)
            skey[j] = s; spid[j] = j;
        } else { skey[j] = -3.0e38f; spid[j] = j; }
    }
    bitonic_desc(skey, spid, 2048, tid, 1024); // stable desc (ties by index)

    for (int j = tid; j < KTOT; j += 1024) {
        const int o = spid[j];
        const float x1 = boxes[(base + o) * 4 + 0], y1 = boxes[(base + o) * 4 + 1];
        const float x2 = boxes[(base + o) * 4 + 2], y2 = boxes[(base + o) * 4 + 3];
        sx1[j] = x1; sy1[j] = y1; sx2[j] = x2; sy2[j] = y2;
        sar[j] = (x2 - x1 + 1.0f) * (y2 - y1 + 1.0f);
        slab[j] = rlab[base + o];
        keep[j] = rval[base + o];
    }
    __syncthreads();

    // exact greedy NMS (matches the reference fori_loop)
    for (int i = 0; i < KTOT; ++i) {
        if (keep[i]) {                          // uniform branch (LDS broadcast)
            const float xi1 = sx1[i], yi1 = sy1[i], xi2 = sx2[i], yi2 = sy2[i], ai = sar[i];
            const int li = slab[i];
            for (int j = i + 1 + tid; j < KTOT; j += 1024) {
                if (keep[j] && slab[j] == li) {
                    float iw = fminf(xi2, sx2[j]) - fmaxf(xi1, sx1[j]) + 1.0f; iw = fmaxf(iw, 0.0f);
                    float ih = fminf(yi2, sy2[j]) - fmaxf(yi1, sy1[j]) + 1.0f; ih = fmaxf(ih, 0.0f);
                    const float inter = iw * ih;
                    const float iou = inter / (ai + sar[j] - inter + 1e-9f);
                    if (iou > NMS_TH) keep[j] = 0;
                }
            }
        }
        __syncthreads();
    }

    // top-100 of fs = keep ? score : -1  (fill with -1 entries in sorted order)
    int* outpos = spid;
    for (int j = tid; j < KTOT; j += 1024) outpos[j] = -1;
    __syncthreads();
    if (tid == 0) {
        int kc = 0;
        for (int j = 0; j < KTOT; ++j) if (keep[j]) { if (kc < POSTK) outpos[j] = kc; kc++; }
        int fc = kc;
        for (int j = 0; j < KTOT && fc < POSTK; ++j) if (!keep[j]) { outpos[j] = fc; fc++; }
    }
    __syncthreads();

    float* obox = out;                       // [N,100,4]
    float* osc  = out + NIMG * POSTK * 4;    // [N,100]
    float* olab = osc + NIMG * POSTK;        // [N,100]
    float* oval = olab + NIMG * POSTK;       // [N,100]
    for (int j = tid; j < KTOT; j += 1024) {
        const int p = outpos[j];
        if (p >= 0) {
            const size_t ob = (size_t)n * POSTK + p;
            obox[ob * 4 + 0] = sx1[j]; obox[ob * 4 + 1] = sy1[j];
            obox[ob * 4 + 2] = sx2[j]; obox[ob * 4 + 3] = sy2[j];
            const float tv = keep[j] ? skey[j] : -1.0f;
            osc[ob]  = tv;
            olab[ob] = (float)slab[j];
            oval[ob] = (tv > 0.0f) ? 1.0f : 0.0f;
        }
    }
}

// ---------------- host launcher ----------------
extern "C" void kernel_launch(void* const* d_in, const int* in_sizes, int n_in,
                              void* d_out, int out_size, void* d_ws, size_t ws_size,
                              hipStream_t stream) {
    (void)in_sizes; (void)n_in; (void)out_size; (void)ws_size;
    // setup_inputs dict order: cls0, gau0, cls1, gau1, cls2, gau2
    const float* cls0 = (const float*)d_in[0];
    const float* gau0 = (const float*)d_in[1];
    const float* cls1 = (const float*)d_in[2];
    const float* gau1 = (const float*)d_in[3];
    const float* cls2 = (const float*)d_in[4];
    const float* gau2 = (const float*)d_in[5];

    char* ws = (char*)d_ws;
    size_t off = 0;
    int*   counters = (int*)(ws + off); off += 256;
    float* candS    = (float*)(ws + off); off += (size_t)12 * CAP * 4;
    int*   candI    = (int*)(ws + off);   off += (size_t)12 * CAP * 4;
    float* topS     = (float*)(ws + off); off += (size_t)12 * TOPPAD * 4;
    int*   topI     = (int*)(ws + off);   off += (size_t)12 * TOPPAD * 4;
    float* boxes    = (float*)(ws + off); off += (size_t)NIMG * KTOT * 4 * 4;
    float* rsc      = (float*)(ws + off); off += (size_t)NIMG * KTOT * 4;
    int*   rlab     = (int*)(ws + off);   off += (size_t)NIMG * KTOT * 4;
    int*   rval     = (int*)(ws + off);   off += (size_t)NIMG * KTOT * 4;

    init_kernel<<<1, 32, 0, stream>>>(counters);

    dim3 g0((160 + SCAN_TH - 1) / SCAN_TH, NCLS, NIMG);
    scan_kernel<160, 256><<<g0, SCAN_B, 0, stream>>>(cls0, gau0, 0, candS, candI, counters);
    dim3 g1((80 + SCAN_TH - 1) / SCAN_TH, NCLS, NIMG);
    scan_kernel<80, 128><<<g1, SCAN_B, 0, stream>>>(cls1, gau1, 1, candS, candI, counters);
    dim3 g2((40 + SCAN_TH - 1) / SCAN_TH, NCLS, NIMG);
    scan_kernel<40, 64><<<g2, SCAN_B, 0, stream>>>(cls2, gau2, 2, candS, candI, counters);

    select_kernel<<<12, 1024, 0, stream>>>(candS, candI, counters, topS, topI);

    solve_kernel<<<(NIMG * KTOT + 255) / 256, 256, 0, stream>>>(
        gau0, gau1, gau2, topS, topI, boxes, rsc, rlab, rval);

    nms_kernel<<<NIMG, 1024, 0, stream>>>(boxes, rsc, rlab, rval, (float*)d_out);
}